// RandomCrop_3917010174291
// MI455X (gfx1250) — compile-verified
//
#include <hip/hip_runtime.h>
#include <hip/hip_bf16.h>
#include <stdint.h>

// ---------------------------------------------------------------------------
// RandomCrop (B=256, C=3, 224x224 -> 192x192, fp32) as a pure TDM DMA copy.
//
// A crop is a strided 2D tile move — exactly what the Tensor Data Mover does.
// Each workgroup (1 wave32) moves one 64-row x 192-col tile of one (b,c)
// plane:
//   TENSOR_LOAD_TO_LDS    global (row stride 224 floats) -> LDS (48 KB)
//   s_wait_tensorcnt 0
//   TENSOR_STORE_FROM_LDS LDS -> global (row stride 192 floats, contiguous)
// Grid = (3 row-blocks, 3 channels, 256 images) = 2304 workgroups; thousands
// of outstanding DMA tiles saturate the 23.3 TB/s HBM (roofline ~9.7 us for
// 2 x 113 MB of traffic). VALU/WMMA are irrelevant for a 0-FLOP op.
// ---------------------------------------------------------------------------

#define CROP_H 192
#define CROP_W 192
#define IN_H   224
#define IN_W   224
#define NB     256
#define NC     3
#define ROWS_PER_BLOCK 64
#define ROWBLOCKS (CROP_H / ROWS_PER_BLOCK)       // 3
#define TILE_BYTES (ROWS_PER_BLOCK * CROP_W * 4)  // 49152

typedef unsigned int v4u __attribute__((ext_vector_type(4)));
typedef int          v4i __attribute__((ext_vector_type(4)));
typedef int          v8i __attribute__((ext_vector_type(8)));

#if defined(__HIP_DEVICE_COMPILE__) && __has_builtin(__builtin_amdgcn_tensor_load_to_lds)
#define USE_TDM 1
#else
#define USE_TDM 0
#endif

// ---- Tensor DMA Descriptor construction (CDNA5 ISA) ----

// Group 0 (128b): [1:0]=count=1, [63:32]=lds_addr, [120:64]=global_addr,
//                 [127:126]=type=2 ("image").
static __device__ __forceinline__ v4u tdm_group0(unsigned lds_addr,
                                                 unsigned long long gaddr) {
    v4u g;
    g.x = 1u;                                      // count=1, user mode
    g.y = lds_addr;                                // LDS byte address
    g.z = (unsigned)(gaddr & 0xFFFFFFFFu);         // global_addr[31:0]
    g.w = (unsigned)((gaddr >> 32) & 0x1FFFFFFu)   // global_addr[56:32]
          | (2u << 30);                            // type=2
    return g;
}

// Group 1 (256b): data_size[17:16], tensor_dim0[79:48], tensor_dim1[111:80],
//                 tile_dim0[127:112], tile_dim1[143:128], tile_dim2[159:144],
//                 tensor_dim0_stride[207:160], tensor_dim1_stride[255:208].
static __device__ __forceinline__ v8i tdm_group1(unsigned tensor_dim0,
                                                 unsigned tensor_dim1,
                                                 unsigned tile_dim0,
                                                 unsigned tile_dim1,
                                                 unsigned long long stride0) {
    v8i g;
    g[0] = (int)(2u << 16);                        // data_size = 4 bytes
    g[1] = (int)((tensor_dim0 & 0xFFFFu) << 16);   // dim0 lo16 (barrier addr = 0)
    g[2] = (int)((tensor_dim0 >> 16) | ((tensor_dim1 & 0xFFFFu) << 16));
    g[3] = (int)((tensor_dim1 >> 16) | ((tile_dim0 & 0xFFFFu) << 16));
    g[4] = (int)(tile_dim1 & 0xFFFFu);             // tile_dim2 = 0 (2D tile)
    g[5] = (int)(stride0 & 0xFFFFFFFFu);           // dim0 stride lo32
    g[6] = (int)((stride0 >> 32) & 0xFFFFu);       // dim0 stride hi16, dim1 stride = 0
    g[7] = 0;
    return g;
}

#if USE_TDM
static __device__ __forceinline__ void tdm_load(v4u g0, v8i g1) {
    v4i z4 = {0, 0, 0, 0};
#if defined(__clang_major__) && (__clang_major__ >= 23)
    v8i z8 = {0, 0, 0, 0, 0, 0, 0, 0};
    __builtin_amdgcn_tensor_load_to_lds(g0, g1, z4, z4, z8, 0);
#else
    __builtin_amdgcn_tensor_load_to_lds(g0, g1, z4, z4, 0);
#endif
}

static __device__ __forceinline__ void tdm_store(v4u g0, v8i g1) {
    v4i z4 = {0, 0, 0, 0};
#if defined(__clang_major__) && (__clang_major__ >= 23)
    v8i z8 = {0, 0, 0, 0, 0, 0, 0, 0};
    __builtin_amdgcn_tensor_store_from_lds(g0, g1, z4, z4, z8, 0);
#else
    __builtin_amdgcn_tensor_store_from_lds(g0, g1, z4, z4, 0);
#endif
}
#endif // USE_TDM

// ---------------------------------------------------------------------------

__global__ void randomcrop_tdm_kernel(const float* x,
                                      const int* idx_h,
                                      const int* idx_w,
                                      const int* training,
                                      float* out) {
#if USE_TDM
    extern __shared__ char tdm_lds[];  // dynamic LDS tile at group-segment offset 0
    (void)tdm_lds;

    // 3D grid: x = row-block (0..2), y = channel (0..2), z = image (0..255).
    const unsigned rb    = blockIdx.x;
    const unsigned plane = blockIdx.z * NC + blockIdx.y;   // b*NC + c
    const unsigned b     = blockIdx.z;

    // Branch-free training/eval select; n_windows = (224-192+1)//2 = 16.
    int ih = idx_h[b];
    int iw = idx_w[b];
    if (training[0] == 0) { ih = 16; iw = 16; }

    const unsigned row0 = (unsigned)ih + rb * ROWS_PER_BLOCK;

    const unsigned long long src_elem =
        (unsigned long long)plane * (IN_H * IN_W) +
        (unsigned long long)row0 * IN_W + (unsigned)iw;
    const unsigned long long dst_elem =
        (unsigned long long)plane * (CROP_H * CROP_W) +
        (unsigned long long)(rb * ROWS_PER_BLOCK) * CROP_W;

    const unsigned long long src_addr = (unsigned long long)(uintptr_t)x   + src_elem * 4ull;
    const unsigned long long dst_addr = (unsigned long long)(uintptr_t)out + dst_elem * 4ull;

    // DMA load: 64 rows x 192 floats, source row stride 224 floats.
    tdm_load(tdm_group0(0u, src_addr),
             tdm_group1(1u << 30, 1u << 30, CROP_W, ROWS_PER_BLOCK, IN_W));

    __builtin_amdgcn_s_wait_tensorcnt(0);

    // DMA store: 64 rows x 192 floats, dest row stride 192 floats.
    tdm_store(tdm_group0(0u, dst_addr),
              tdm_group1(1u << 30, 1u << 30, CROP_W, ROWS_PER_BLOCK, CROP_W));
    // S_ENDPGM implicitly waits for all counters (incl. TENSORcnt).
#else
    // Fallback: vectorized copy (output float4-aligned; source only 4B-aligned
    // because of idx_w, so gather with scalar dword loads).
    unsigned long long tid = (unsigned long long)blockIdx.x * blockDim.x + threadIdx.x;
    const unsigned n_vec4 = (NB * NC * CROP_H * CROP_W) / 4;
    if (tid >= n_vec4) return;

    unsigned v  = (unsigned)tid;
    unsigned w4 = v % (CROP_W / 4);
    unsigned t  = v / (CROP_W / 4);
    unsigned h  = t % CROP_H;
    unsigned pl = t / CROP_H;
    unsigned b  = pl / NC;

    int ih = idx_h[b];
    int iw = idx_w[b];
    if (training[0] == 0) { ih = 16; iw = 16; }

    const float* src = x + (unsigned long long)pl * (IN_H * IN_W)
                         + (unsigned long long)(ih + h) * IN_W + iw + w4 * 4;
    float4 r;
    r.x = src[0]; r.y = src[1]; r.z = src[2]; r.w = src[3];
    ((float4*)out)[v] = r;
#endif
}

extern "C" void kernel_launch(void* const* d_in, const int* in_sizes, int n_in,
                              void* d_out, int out_size, void* d_ws, size_t ws_size,
                              hipStream_t stream) {
    const float* x        = (const float*)d_in[0];
    const int*   idx_h    = (const int*)d_in[1];
    const int*   idx_w    = (const int*)d_in[2];
    const int*   training = (const int*)d_in[3];
    float*       out      = (float*)d_out;

#if 1
    // TDM path: one single-wave workgroup per 64-row tile of each (b,c) plane.
    dim3 grid(ROWBLOCKS, NC, NB);   // 3 x 3 x 256 = 2304 workgroups
    dim3 block(32);                 // one wave32
    randomcrop_tdm_kernel<<<grid, block, TILE_BYTES, stream>>>(x, idx_h, idx_w, training, out);
#else
    const unsigned n_vec4 = (NB * NC * CROP_H * CROP_W) / 4;
    dim3 grid((n_vec4 + 255) / 256);
    dim3 block(256);
    randomcrop_tdm_kernel<<<grid, block, 0, stream>>>(x, idx_h, idx_w, training, out);
#endif
}